// HungarianMatcher_87668872446510
// MI455X (gfx1250) — compile-verified
//
#include <hip/hip_runtime.h>
#include <hip/hip_bf16.h>

typedef __attribute__((ext_vector_type(2))) float v2f;
typedef __attribute__((ext_vector_type(8))) float v8f;

#define T_TGT   320
#define P_PTS   72
#define TGT_W   149   // 5 + 2*72
#define NQ      1600
#define YS_OFF  77    // 5 + P
#define XS_OFF  5

// ---------------------------------------------------------------------------
// Prep: vcount[t] = #valid points of target t; minc = min over t.
// weights[t] = sqrt(sum(valid)/vcount[t]) / max(...) == sqrt(minc/vcount[t])
// (global totals cancel). Single block, deterministic.
// ---------------------------------------------------------------------------
__global__ void hm_prep_kernel(const float* __restrict__ tgt,
                               int* __restrict__ vcount,
                               int* __restrict__ minc) {
    __shared__ int smin;
    if (threadIdx.x == 0) smin = 0x7fffffff;
    __syncthreads();
    int t = threadIdx.x;
    if (t < T_TGT) {
        int c = 0;
        const float* xs = tgt + t * TGT_W + XS_OFF;
        for (int p = 0; p < P_PTS; ++p) c += (xs[p] >= 0.0f) ? 1 : 0;
        vcount[t] = c;
        atomicMin(&smin, c);
    }
    __syncthreads();
    if (threadIdx.x == 0) *minc = smin;
}

// ---------------------------------------------------------------------------
// Main: one wave handles (target t) x (16 queries). The cubic evaluation
// out_xs = [1,y,y^2,y^3] . b[0..3] is a K=4 contraction -> one
// v_wmma_f32_16x16x4_f32 per 16-point tile (4 full tiles + 1 tail tile cover
// P=72 padded to 80). In the tail tile the valid/padded split falls exactly
// on the half-wave boundary (hi=0: p=64..71 valid, hi=1: p=72..79 padded),
// so the epilogue stays contiguous-b128 loads with one uniform half-mask.
// ---------------------------------------------------------------------------
__global__ __launch_bounds__(256) void hm_cost_kernel(
    const float* __restrict__ logits,    // (1600, 2)
    const float* __restrict__ curves,    // (1600, 8)
    const float* __restrict__ tgt,       // (320, 149)
    const int*   __restrict__ tgt_ids,   // (320,)
    const int*   __restrict__ vcount,    // (320,)
    const int*   __restrict__ minc,      // scalar
    float* __restrict__ out)             // (1600, 320) row-major in q
{
    const int wave   = (blockIdx.x * blockDim.x + threadIdx.x) >> 5;
    const int lane   = threadIdx.x & 31;
    const int col    = lane & 15;        // query column / row M of A-tile
    const int hi     = lane >> 4;        // half-wave: K-split for A/B operands
    const int t      = wave % T_TGT;
    const int q_base = (wave / T_TGT) * 16;
    const int q      = q_base + col;

    const float* bc   = curves + q * 8;
    const float* trow = tgt + t * TGT_W;

    // B matrix 4x16 (K-major per lane): lanes 0-15 hold K=0,1; lanes 16-31 K=2,3
    v2f Bm;
    Bm.x = bc[2 * hi + 0];
    Bm.y = bc[2 * hi + 1];

    float acc = 0.0f;

    // ---- Tiles 0..3: p in [0,64), always in range -> no compares, no masks
    #pragma unroll
    for (int i = 0; i < 4; ++i) {
        const int p0 = i * 16;

        const float y  = trow[YS_OFF + p0 + col];
        const float y2 = y * y;
        v2f Am;
        Am.x = hi ? y2 : 1.0f;
        Am.y = hi ? y2 * y : y;

        v8f c0 = {};
        v8f d = __builtin_amdgcn_wmma_f32_16x16x4_f32(
            false, Am, false, Bm, (short)0, c0, false, false);

        const float* xsp = trow + (XS_OFF + p0 + 8 * hi);  // 8 consecutive
        #pragma unroll
        for (int r = 0; r < 8; ++r) {
            const float xs = xsp[r];
            const float m  = (xs >= 0.0f) ? 1.0f : 0.0f;
            acc += m * fabsf(xs - d[r]);
        }
    }

    // ---- Tail tile: p in [64,80). hi=0 rows are all valid (64..71), hi=1
    // rows are all padding (72..79) -> uniform half-mask, contiguous loads.
    {
        const int   pa  = 64 + col;
        const int   pac = (pa < P_PTS) ? pa : 0;     // single cndmask on index
        float y = trow[YS_OFF + pac];
        y = (pa < P_PTS) ? y : 0.0f;
        const float y2 = y * y;
        v2f Am;
        Am.x = hi ? y2 : 1.0f;
        Am.y = hi ? y2 * y : y;

        v8f c0 = {};
        v8f d = __builtin_amdgcn_wmma_f32_16x16x4_f32(
            false, Am, false, Bm, (short)0, c0, false, false);

        const float hmask = hi ? 0.0f : 1.0f;        // hi half contributes 0
        const float* xsp = trow + (XS_OFF + 64);     // p=64..71, in range
        #pragma unroll
        for (int r = 0; r < 8; ++r) {
            const float xs = xsp[r];
            const float m  = (xs >= 0.0f) ? hmask : 0.0f;
            acc += m * fabsf(xs - d[r]);
        }
    }

    // Fold the two half-wave partial sums (rows 0-7 vs 8-15 of each tile).
    acc += __shfl_xor(acc, 16, 32);

    // weights[t] = sqrt(min_vcount / vcount[t])
    const float w = sqrtf((float)(*minc) / (float)vcount[t]);
    const float cost_polys = acc * w;

    // cost_class = -softmax(logits[q])[tgt_ids[t]]   (2 classes)
    const float l0 = logits[q * 2 + 0];
    const float l1 = logits[q * 2 + 1];
    const float mx = fmaxf(l0, l1);
    const float e0 = __expf(l0 - mx);
    const float e1 = __expf(l1 - mx);
    const int   id = tgt_ids[t];
    const float cost_class = -((id == 0) ? e0 : e1) / (e0 + e1);

    // lower / upper L1 endpoint costs
    const float cl = 0.5f * (fabsf(bc[4] - trow[1]) + fabsf(bc[6] - trow[3]));
    const float cu = 0.5f * (fabsf(bc[5] - trow[2]) + fabsf(bc[7] - trow[4]));

    if (hi == 0) {
        out[q * T_TGT + t] = cost_class + 0.1f * cost_polys + cl + cu;
    }
}

extern "C" void kernel_launch(void* const* d_in, const int* in_sizes, int n_in,
                              void* d_out, int out_size, void* d_ws, size_t ws_size,
                              hipStream_t stream) {
    const float* logits  = (const float*)d_in[0];   // (16,100,2)
    const float* curves  = (const float*)d_in[1];   // (16,100,8)
    const float* tgt     = (const float*)d_in[2];   // (320,149)
    const int*   tgt_ids = (const int*)d_in[3];     // (320,)
    float* out = (float*)d_out;

    int* vcount = (int*)d_ws;          // 320 ints
    int* minc   = vcount + T_TGT;      // 1 int

    hm_prep_kernel<<<1, T_TGT, 0, stream>>>(tgt, vcount, minc);

    // 320 targets x 100 query-blocks = 32000 waves; 8 waves per 256-thread block
    const int total_waves = T_TGT * (NQ / 16);
    const int blocks = total_waves / 8;
    hm_cost_kernel<<<blocks, 256, 0, stream>>>(logits, curves, tgt, tgt_ids,
                                               vcount, minc, out);
}